// GNNFeatureExtractor_69002944577974
// MI455X (gfx1250) — compile-verified
//
#include <hip/hip_runtime.h>

#define NN 50000
#define NE 800000
#define TM 3125   // NN / 16

typedef __attribute__((ext_vector_type(2))) float v2f;
typedef __attribute__((ext_vector_type(8))) float v8f;
typedef __attribute__((ext_vector_type(4))) unsigned int u32x4;
typedef __attribute__((ext_vector_type(4))) int i32x4;
typedef __attribute__((ext_vector_type(8))) int i32x8;

__device__ __forceinline__ void atomAddF(float* p, float v) {
    __hip_atomic_fetch_add(p, v, __ATOMIC_RELAXED, __HIP_MEMORY_SCOPE_AGENT);
}

__global__ void k_zero(float* __restrict__ p, int n) {
    for (int i = blockIdx.x * blockDim.x + threadIdx.x; i < n; i += gridDim.x * blockDim.x)
        p[i] = 0.0f;
}

__global__ void k_degree(const int* __restrict__ dst, float* __restrict__ deg) {
    for (int i = blockIdx.x * blockDim.x + threadIdx.x; i < NE; i += gridDim.x * blockDim.x)
        atomAddF(&deg[dst[i]], 1.0f);
}

__global__ void k_invdeg(float* __restrict__ deg) {
    int i = blockIdx.x * blockDim.x + threadIdx.x;
    if (i < NN) deg[i] = 1.0f / fmaxf(deg[i], 1.0f);
}

// One wave per edge (F=128) or per 2 edges (F=64): lane-groups gather one full
// feature row as float4s (coalesced 512B/256B) and scatter-add via L2 f32 atomics.
template<int F>
__global__ void k_scatter(const int* __restrict__ src, const int* __restrict__ dst,
                          const float* __restrict__ in, float* __restrict__ agg) {
    constexpr int QPE = F / 4;        // float4 quads per edge row (32 or 16)
    constexpr int EPW = 32 / QPE;     // edges handled per wave (1 or 2)
    const int nWaves = (gridDim.x * blockDim.x) >> 5;
    const int gwave  = (blockIdx.x * blockDim.x + threadIdx.x) >> 5;
    const int lane   = threadIdx.x & 31;
    const int q      = lane & (QPE - 1);
    const int sub    = lane >> ((F == 128) ? 5 : 4);   // 0, or 0/1 when EPW==2
    for (int e0 = gwave * EPW; e0 < NE; e0 += nWaves * EPW) {
        int e = e0 + sub;                              // NE is even: always in range
        int s = src[e];
        int d = dst[e];
        float4 v = *reinterpret_cast<const float4*>(in + (size_t)s * F + q * 4);
        float* o = agg + (size_t)d * F + q * 4;
        atomAddF(o + 0, v.x);
        atomAddF(o + 1, v.y);
        atomAddF(o + 2, v.z);
        atomAddF(o + 3, v.w);
    }
}

#if defined(__HIP_DEVICE_COMPILE__)
// TDM: load a 2-D tile (nRows x 16 f32, row stride = rowStride elements) into LDS.
// D# layout per CDNA5 ISA §8.3/8.4: group0 = {count/type/addr/lds}, group1 = dims.
__device__ __forceinline__ void tdm_load_tile(const float* gptr, unsigned ldsOff,
                                              int rowLen, int nRows, int rowStride) {
    unsigned long long ga = (unsigned long long)(uintptr_t)gptr;
    unsigned td0 = (unsigned)rowLen;        // tensor_dim0
    unsigned td1 = (unsigned)nRows;         // tensor_dim1
    u32x4 g0;
    g0.x = 1u;                                                 // count=1, user mode
    g0.y = ldsOff;                                             // lds_addr
    g0.z = (unsigned)(ga & 0xFFFFFFFFu);                       // global_addr[31:0]
    g0.w = (unsigned)((ga >> 32) & 0x1FFFFFFu) | (2u << 30);   // addr[56:32] | type=2
    i32x8 g1;
    g1[0] = (int)(2u << 16);                                   // data_size = 4B
    g1[1] = (int)((td0 & 0xFFFFu) << 16);                      // tensor_dim0 lo16
    g1[2] = (int)((td0 >> 16) | ((td1 & 0xFFFFu) << 16));      // td0 hi16 | td1 lo16
    g1[3] = (int)((td1 >> 16) | (16u << 16));                  // td1 hi16 | tile_dim0=16
    g1[4] = (int)(td1 & 0xFFFFu);                              // tile_dim1 = nRows
    g1[5] = (int)rowStride;                                    // tensor_dim0_stride lo32
    g1[6] = 0;
    g1[7] = 0;
    i32x4 gz = {0, 0, 0, 0};
#if __clang_major__ >= 23
    i32x8 gz8 = {0, 0, 0, 0, 0, 0, 0, 0};
    __builtin_amdgcn_tensor_load_to_lds(g0, g1, gz, gz, gz8, 0);
#else
    __builtin_amdgcn_tensor_load_to_lds(g0, g1, gz, gz, 0);
#endif
}
#endif

// y = (A1 * inv_deg) @ Wl + bl [+ A2 @ Wr]; optional BN column stats.
// Block = 8 waves sharing one 16-column weight slice (tn = blockIdx.x), staged in
// LDS by wave 0 via the Tensor Data Mover; each wave owns one 16x16 output tile.
template<bool STATS>
__global__ void __launch_bounds__(256)
k_sage_gemm(const float* __restrict__ A1, const float* __restrict__ inv_deg,
            const float* __restrict__ A2,
            const float* __restrict__ Wl, const float* __restrict__ bl,
            const float* __restrict__ Wr,
            float* __restrict__ y,
            float* __restrict__ colsum, float* __restrict__ colsq,
            int din, int dout) {
    __shared__ float bsh[4096];               // up to 16KB: [Wl tile | Wr tile]
    const int tn   = blockIdx.x;              // output column tile
    const int tmB  = blockIdx.y << 3;         // first row tile of this block
    const int wave = threadIdx.x >> 5;
    const int lane = threadIdx.x & 31;
    const int hl   = lane >> 4;
    const int l16  = lane & 15;

#if defined(__HIP_DEVICE_COMPILE__)
    if (threadIdx.x < 32) {                   // wave 0 issues the TDM transfers
        unsigned ldsBase = (unsigned)(uintptr_t)&bsh[0];
        tdm_load_tile(Wl + tn * 16, ldsBase, dout, din, dout);
        if (A2)
            tdm_load_tile(Wr + tn * 16, ldsBase + (unsigned)din * 64u, dout, din, dout);
        __builtin_amdgcn_s_wait_tensorcnt(0);
    }
#else
    for (int i = threadIdx.x; i < din * 16; i += blockDim.x) {
        bsh[i] = Wl[(size_t)(i / 16) * dout + tn * 16 + (i & 15)];
        if (A2) bsh[din * 16 + i] = Wr[(size_t)(i / 16) * dout + tn * 16 + (i & 15)];
    }
#endif
    __syncthreads();

    const int tm = tmB + wave;
    if (tm >= TM) return;                     // wave-uniform: EXEC all-1s below

    const int rowA = tm * 16 + l16;           // A fragment: lane = M row
    const int colB = tn * 16 + l16;           // B fragment: lane = N col
    v8f c = {0.f, 0.f, 0.f, 0.f, 0.f, 0.f, 0.f, 0.f};
    const float idg = inv_deg ? inv_deg[rowA] : 1.0f;

    {   // neighbor term: (agg * inv_deg) @ Wl
        const float* Ar = A1 + (size_t)rowA * din;
        #pragma unroll 4
        for (int k0 = 0; k0 < din; k0 += 4) {
            const int kb = k0 + hl * 2;
            v2f a, b;
            a.x = Ar[kb] * idg;
            a.y = Ar[kb + 1] * idg;
            b.x = bsh[kb * 16 + l16];
            b.y = bsh[kb * 16 + 16 + l16];
            c = __builtin_amdgcn_wmma_f32_16x16x4_f32(false, a, false, b, (short)0, c,
                                                      false, false);
        }
    }
    if (A2) {  // root term: x @ Wr
        const float* Ar = A2 + (size_t)rowA * din;
        const float* Br = bsh + din * 16;
        #pragma unroll 4
        for (int k0 = 0; k0 < din; k0 += 4) {
            const int kb = k0 + hl * 2;
            v2f a, b;
            a.x = Ar[kb];
            a.y = Ar[kb + 1];
            b.x = Br[kb * 16 + l16];
            b.y = Br[kb * 16 + 16 + l16];
            c = __builtin_amdgcn_wmma_f32_16x16x4_f32(false, a, false, b, (short)0, c,
                                                      false, false);
        }
    }

    const float bias = bl[colB];
    float s = 0.f, ss = 0.f;
    #pragma unroll
    for (int r = 0; r < 8; ++r) {             // C/D layout: VGPR r -> M = r + hl*8, N = l16
        float v = c[r] + bias;
        int row = tm * 16 + hl * 8 + r;
        y[(size_t)row * dout + colB] = v;
        s  += v;
        ss += v * v;
    }
    if (STATS) {
        atomAddF(&colsum[colB], s);
        atomAddF(&colsq[colB], ss);
    }
}

__global__ void k_bn_finalize(const float* __restrict__ colsum, const float* __restrict__ colsq,
                              const float* __restrict__ g, const float* __restrict__ be,
                              float* __restrict__ scale, float* __restrict__ shift, int dout) {
    int f = blockIdx.x * blockDim.x + threadIdx.x;
    if (f < dout) {
        const float invN = 1.0f / (float)NN;
        float m   = colsum[f] * invN;
        float var = colsq[f] * invN - m * m;
        float sc  = g[f] * rsqrtf(var + 1e-5f);
        scale[f] = sc;
        shift[f] = be[f] - m * sc;
    }
}

__global__ void k_bn_relu(const float* __restrict__ y, const float* __restrict__ scale,
                          const float* __restrict__ shift, float* __restrict__ h,
                          int mask, int total) {
    for (int i = blockIdx.x * blockDim.x + threadIdx.x; i < total; i += gridDim.x * blockDim.x) {
        int f = i & mask;
        h[i] = fmaxf(fmaf(y[i], scale[f], shift[f]), 0.0f);
    }
}

extern "C" void kernel_launch(void* const* d_in, const int* in_sizes, int n_in,
                              void* d_out, int out_size, void* d_ws, size_t ws_size,
                              hipStream_t stream) {
    const float* x   = (const float*)d_in[0];
    const int*   ei  = (const int*)d_in[1];
    const int*   src = ei;
    const int*   dst = ei + NE;
    const float* W1l = (const float*)d_in[2];
    const float* b1  = (const float*)d_in[3];
    const float* W1r = (const float*)d_in[4];
    const float* g1  = (const float*)d_in[5];
    const float* be1 = (const float*)d_in[6];
    const float* W2l = (const float*)d_in[7];
    const float* b2  = (const float*)d_in[8];
    const float* W2r = (const float*)d_in[9];
    const float* g2  = (const float*)d_in[10];
    const float* be2 = (const float*)d_in[11];
    const float* W3l = (const float*)d_in[12];
    const float* b3  = (const float*)d_in[13];
    const float* W3r = (const float*)d_in[14];
    const float* g3  = (const float*)d_in[15];
    const float* be3 = (const float*)d_in[16];
    const float* Wp  = (const float*)d_in[17];
    const float* bp  = (const float*)d_in[18];

    float* P      = (float*)d_ws;            // 50000*256  feature buffer (h1/h2/h3)
    float* Qb     = P + (size_t)NN * 256;    // 50000*256  pre-BN y buffer
    float* R      = Qb + (size_t)NN * 256;   // 50000*128  aggregation buffer
    float* invdeg = R + (size_t)NN * 128;    // 50000
    float* colsum = invdeg + NN;             // 256
    float* colsq  = colsum + 256;            // 256
    float* scl    = colsq + 256;             // 256
    float* sft    = scl + 256;               // 256

    const dim3 B(256);
    const int  gy = (TM + 7) / 8;            // 391 row-tile groups of 8 waves

    // degrees -> inverse clipped degree (1/max(deg,1))
    k_zero<<<256, B, 0, stream>>>(invdeg, NN);
    k_degree<<<1024, B, 0, stream>>>(dst, invdeg);
    k_invdeg<<<(NN + 255) / 256, B, 0, stream>>>(invdeg);

    // ---------------- Layer 1: 128 -> 64 ----------------
    k_zero<<<2048, B, 0, stream>>>(R, NN * 128);
    k_zero<<<1, B, 0, stream>>>(colsum, 512);
    k_scatter<128><<<8192, B, 0, stream>>>(src, dst, x, R);
    k_sage_gemm<true><<<dim3(64 / 16, gy), B, 0, stream>>>(R, invdeg, x, W1l, b1, W1r,
                                                           Qb, colsum, colsq, 128, 64);
    k_bn_finalize<<<1, B, 0, stream>>>(colsum, colsq, g1, be1, scl, sft, 64);
    k_bn_relu<<<2048, B, 0, stream>>>(Qb, scl, sft, P, 63, NN * 64);   // h1 -> P

    // ---------------- Layer 2: 64 -> 128 ----------------
    k_zero<<<2048, B, 0, stream>>>(R, NN * 64);
    k_zero<<<1, B, 0, stream>>>(colsum, 512);
    k_scatter<64><<<8192, B, 0, stream>>>(src, dst, P, R);
    k_sage_gemm<true><<<dim3(128 / 16, gy), B, 0, stream>>>(R, invdeg, P, W2l, b2, W2r,
                                                            Qb, colsum, colsq, 64, 128);
    k_bn_finalize<<<1, B, 0, stream>>>(colsum, colsq, g2, be2, scl, sft, 128);
    k_bn_relu<<<2048, B, 0, stream>>>(Qb, scl, sft, P, 127, NN * 128); // h2 -> P

    // ---------------- Layer 3: 128 -> 256 ----------------
    k_zero<<<2048, B, 0, stream>>>(R, NN * 128);
    k_zero<<<1, B, 0, stream>>>(colsum, 512);
    k_scatter<128><<<8192, B, 0, stream>>>(src, dst, P, R);
    k_sage_gemm<true><<<dim3(256 / 16, gy), B, 0, stream>>>(R, invdeg, P, W3l, b3, W3r,
                                                            Qb, colsum, colsq, 128, 256);
    k_bn_finalize<<<1, B, 0, stream>>>(colsum, colsq, g3, be3, scl, sft, 256);
    k_bn_relu<<<4096, B, 0, stream>>>(Qb, scl, sft, P, 255, NN * 256); // h3 -> P

    // ---------------- Projection: 256 -> 128 ----------------
    k_sage_gemm<false><<<dim3(128 / 16, gy), B, 0, stream>>>(P, nullptr, nullptr, Wp, bp,
                                                             nullptr, (float*)d_out,
                                                             nullptr, nullptr, 256, 128);
}